// EdgeAttentionAggregator_48430051230127
// MI455X (gfx1250) — compile-verified
//
#include <hip/hip_runtime.h>
#include <math.h>

// EdgeAttentionAggregator for MI455X (gfx1250, wave32, WMMA f32 16x16x4)
// B=8, N=128, E=2, DM=DN=128, H=4. One workgroup per (b, r) pair.
// Tile staging uses CDNA5 async global->LDS loads (ASYNCcnt path).

typedef __attribute__((ext_vector_type(2))) float v2f;
typedef __attribute__((ext_vector_type(8))) float v8f;

#define NN      128           // N
#define NE      2             // E
#define DMC     128           // DM
#define NH      4             // H
#define ROWS    256           // N*E rows per (b,r) tile
#define LDS_S   132           // padded LDS row stride (floats): 4*j mod 64 bank rotation, 16B aligned
#define SM_MSGS (ROWS * LDS_S)   // 33792 floats
#define SM_WGT  (ROWS * 16)      // 4096 floats (scores -> weights, h padded to 16)
#define SM_WM   (16 * DMC)       // 2048 floats (W_msg zero-padded to 16 rows)
#define SM_T    16
#define SMEM_FLOATS (SM_MSGS + SM_WGT + SM_WM + SM_T)  // 39952 floats = 159808 B

__launch_bounds__(256, 2)
__global__ void edge_attn_kernel(const float* __restrict__ edge_msgs,
                                 const float* __restrict__ node_states,
                                 const float* __restrict__ W,
                                 const float* __restrict__ bvec,
                                 const int*   __restrict__ edges,
                                 float*       __restrict__ out)
{
    extern __shared__ float smem[];
    float* lds_msgs = smem;                  // [ROWS][LDS_S]   LDS byte base 0 (no static LDS)
    float* lds_wgt  = smem + SM_MSGS;        // [ROWS][16]
    float* lds_wm   = lds_wgt + SM_WGT;      // [16][DMC]
    float* lds_t    = lds_wm  + SM_WM;       // [4] node term + bias

    const int tid = threadIdx.x;
    const int bb  = blockIdx.x >> 7;         // batch
    const int r   = blockIdx.x & 127;        // receiver node

    // ---------------- Preamble: stage tile + weights into LDS ----------------
    // edge_msgs rows (s,e) for this (b,r): 256 rows x 128 f32.
    // CDNA5 async copy: 16B per lane straight into LDS, no VGPR staging.
    // Dynamic LDS base is 0 (no static __shared__), so raw byte offsets are valid VDST.
    for (int idx = tid; idx < ROWS * 32; idx += 256) {
        const int j  = idx >> 5;             // row 0..255  (j = 2*s + e)
        const int c4 = (idx & 31) << 2;      // float col
        const int s  = j >> 1, e = j & 1;
        const float* gp = edge_msgs +
            ((((size_t)bb * NN + s) * NN + r) * NE + e) * DMC + c4;
        const unsigned lds_off = (unsigned)((j * LDS_S + c4) * sizeof(float));
        asm volatile("global_load_async_to_lds_b128 %0, %1, off"
                     :: "v"(lds_off), "v"(gp) : "memory");
    }
    // Overlap with the async DMA: W_msg copy (zero-padded to 16 rows), wgt zero,
    // and the node-term dot product.
    for (int idx = tid; idx < 512; idx += 256) {          // 512 float4s
        const int h  = idx >> 5;
        const int c4 = (idx & 31) << 2;
        float4 v = make_float4(0.f, 0.f, 0.f, 0.f);
        if (h < NH) v = *(const float4*)(W + h * 256 + c4);
        *(float4*)(lds_wm + h * DMC + c4) = v;
    }
    for (int idx = tid; idx < SM_WGT; idx += 256) lds_wgt[idx] = 0.0f;
    if (tid < NH) {
        const float* ns = node_states + ((size_t)bb * NN + r) * DMC;
        const float* wn = W + tid * 256 + DMC;
        float t = bvec[tid];
        #pragma unroll 8
        for (int c = 0; c < DMC; ++c) t = fmaf(ns[c], wn[c], t);
        lds_t[tid] = t;
    }
    asm volatile("s_wait_asynccnt 0x0" ::: "memory");   // async tile complete
    __syncthreads();

    const int wv   = tid >> 5;               // wave id 0..7
    const int lane = tid & 31;
    const int lh   = lane & 15;              // M/N index within fragment
    const int kp   = (lane >> 4) << 1;       // K sub-offset: 0 or 2

    // ---------------- Phase 1: scores  D[se,h] = msgs x W_msg^T -------------
    for (int mt = 2 * wv; mt < 2 * wv + 2; ++mt) {
        const int m0 = mt << 4;
        const float* arow = lds_msgs + (m0 + lh) * LDS_S;
        const float* brow = lds_wm + lh * DMC;
        v8f acc = {};
        #pragma unroll 8
        for (int kk = 0; kk < 32; ++kk) {
            const int cb = (kk << 2) + kp;
            const v2f a = *(const v2f*)(arow + cb);   // A[m=lh, k=cb..cb+1]
            const v2f b = *(const v2f*)(brow + cb);   // B[k, n=lh] = W[n, c]
            acc = __builtin_amdgcn_wmma_f32_16x16x4_f32(
                false, a, false, b, (short)0, acc, false, false);
        }
        // ELU(dot + node_term + bias_b) -> score LDS (only h<4 columns real)
        if (lh < NH) {
            const float tv = lds_t[lh];
            const int jb = m0 + ((lane >> 4) << 3);   // D rows: vr + 8*(lane/16)
            #pragma unroll
            for (int vr = 0; vr < 8; ++vr) {
                const float x = acc[vr] + tv;
                lds_wgt[(jb + vr) * 16 + lh] = (x > 0.f) ? x : expm1f(x);
            }
        }
    }
    __syncthreads();

    // ---------------- Softmax over s per (e,h): one wave per combo ----------
    {
        const int e = wv & 1;
        const int h = wv >> 1;                // 0..3
        float vals[4];
        float vmax = -INFINITY;
        #pragma unroll
        for (int i = 0; i < 4; ++i) {
            const int s = lane + (i << 5);
            const int j = (s << 1) | e;
            const int eg = edges[(((size_t)bb * NN + s) * NN + r) * (NE + 1) + 1 + e];
            const float bias = (1.0f - (float)eg) * (-1.0e6f);
            const float v = lds_wgt[j * 16 + h] + bias;
            vals[i] = v;
            vmax = fmaxf(vmax, v);
        }
        #pragma unroll
        for (int off = 16; off >= 1; off >>= 1)
            vmax = fmaxf(vmax, __shfl_xor(vmax, off, 32));
        float sum = 0.f;
        #pragma unroll
        for (int i = 0; i < 4; ++i) { vals[i] = expf(vals[i] - vmax); sum += vals[i]; }
        #pragma unroll
        for (int off = 16; off >= 1; off >>= 1)
            sum += __shfl_xor(sum, off, 32);
        const float inv = 1.0f / sum;
        #pragma unroll
        for (int i = 0; i < 4; ++i) {
            const int s = lane + (i << 5);
            const int j = (s << 1) | e;
            lds_wgt[j * 16 + h] = vals[i] * inv;
        }
    }
    __syncthreads();

    // ---------------- Phase 2: out[h,c] = wgt^T x msgs ----------------------
    {
        const int c0 = wv << 4;               // 16-wide c tile per wave
        v8f acc = {};
        #pragma unroll 8
        for (int kk = 0; kk < 64; ++kk) {
            const int jb = (kk << 2) + kp;    // K rows jb, jb+1
            v2f a, b;
            a.x = lds_wgt[jb * 16 + lh];          // A[m=h(lh), k=jb]   (rows h>=4 are 0)
            a.y = lds_wgt[(jb + 1) * 16 + lh];
            b.x = lds_msgs[jb * LDS_S + c0 + lh]; // B[k=jb, n] = msgs[jb, c0+n]
            b.y = lds_msgs[(jb + 1) * LDS_S + c0 + lh];
            acc = __builtin_amdgcn_wmma_f32_16x16x4_f32(
                false, a, false, b, (short)0, acc, false, false);
        }
        // D rows 0..3 (lanes 0..15) are h=0..3; col = c0 + lane
        if (lane < 16) {
            float* op = out + (((size_t)bb * NN + r) * (NH * DMC)) + c0 + lane;
            #pragma unroll
            for (int vr = 0; vr < NH; ++vr) op[vr * DMC] = acc[vr];
        }
    }
}

extern "C" void kernel_launch(void* const* d_in, const int* in_sizes, int n_in,
                              void* d_out, int out_size, void* d_ws, size_t ws_size,
                              hipStream_t stream) {
    (void)in_sizes; (void)n_in; (void)out_size; (void)d_ws; (void)ws_size;
    const float* edge_msgs   = (const float*)d_in[0];
    const float* node_states = (const float*)d_in[1];
    const float* W           = (const float*)d_in[2];
    const float* bvec        = (const float*)d_in[3];
    const int*   edges       = (const int*)d_in[4];
    float* out = (float*)d_out;

    const size_t smem_bytes = (size_t)SMEM_FLOATS * sizeof(float);  // 159808 B
    edge_attn_kernel<<<dim3(8 * NN), dim3(256), smem_bytes, stream>>>(
        edge_msgs, node_states, W, bvec, edges, out);
}